// multi_grid_encoder_72241349919099
// MI455X (gfx1250) — compile-verified
//
#include <hip/hip_runtime.h>
#include <hip/hip_bf16.h>
#include <math.h>

typedef __attribute__((ext_vector_type(2))) float v2f;
typedef __attribute__((ext_vector_type(8))) float v8f;

#define NC    65536
#define NHN   7
#define NVV   4
#define EOUT  16

// ---- LDS layout (floats) ----
// wT   : [0 .. 4127]  W transposed [f(16)][k(256)], row stride 258 (bank-conflict-free)
// per wave w (0..7), base = 4128 + w*7600:
//   xgI : [0    .. 2047]  gathered x, [c(2)][h/2(4)][col(128)][h&1(2)] (pairs contiguous)
//   ndT : [2048 .. 2367]  weights,    [c(2)][os(16)][h(10: 8 + pad)]   (pairs contiguous)
//   Y   : [2368 .. 7551]  intermediate, c-stride 2592, row stride 160, skew 2*(b*4+v)
//   dS  : [7552 .. 7567]  broadcast distances  [c(2)][8]
//   mS  : [7568 .. 7583]  broadcast masks      [c(2)][8]
//   iS  : [7584 .. 7599]  broadcast adjc (int) [c(2)][8]
#define W_STRIDE    258
#define ND_STRIDE   10
#define Y_CSTRIDE   2592
#define Y_RSTRIDE   160
#define WAVE_LDS    7600
#define LDS_FLOATS  (4128 + 8 * WAVE_LDS)

__global__ __launch_bounds__(256) void
multi_grid_encoder_kernel(const float* __restrict__ x,
                          const float* __restrict__ coords,
                          const float* __restrict__ sigma,
                          const float* __restrict__ dist_offsets,
                          const float* __restrict__ Wg,
                          const float* __restrict__ b_out,
                          const int*   __restrict__ adjc,
                          const int*   __restrict__ nh_mask,
                          float*       __restrict__ out)
{
    extern __shared__ float lds[];

    const int tid    = threadIdx.x;
    const int wave   = tid >> 5;
    const int lane   = tid & 31;
    const int lane16 = lane & 15;
    const int hi     = lane >> 4;           // half-wave select

    float* wS  = lds;
    float* wb  = lds + 4128 + wave * WAVE_LDS;
    float* xgP = wb;
    float* ndP = wb + 2048;
    float* yP  = wb + 2368;
    float* dS  = wb + 7552;
    float* mS  = wb + 7568;
    int*   iS  = (int*)(wb + 7584);

    // ---- cooperative transposed load of W into LDS: wT[f][k] = W[k*16+f] ----
    {
        const float4* src = (const float4*)Wg;
        #pragma unroll
        for (int it = 0; it < 4; ++it) {
            const int i4 = tid + it * 256;
            const float4 v = src[i4];
            const int k  = i4 >> 2;
            const int f0 = (i4 & 3) * 4;
            wS[(f0 + 0) * W_STRIDE + k] = v.x;
            wS[(f0 + 1) * W_STRIDE + k] = v.y;
            wS[(f0 + 2) * W_STRIDE + k] = v.z;
            wS[(f0 + 3) * W_STRIDE + k] = v.w;
        }
    }

    const int pairIdx = blockIdx.x * 8 + wave;      // one wave = 2 cells
    const int n0      = pairIdx * 2;
    const int c       = hi;

    // ---- stage 0a: distances on 7 lanes per cell (hardware transcendentals) ----
    if (lane16 < NHN) {
        const int h = lane16;
        const int n = n0 + c;
        const float lat_c = coords[n];
        const float lon_c = coords[NC + n];
        const int   m     = adjc[n * NHN + h];
        const float lat_n = coords[m];
        const float lon_n = coords[NC + m];
        float cosd = __sinf(lat_c) * __sinf(lat_n)
                   + __cosf(lat_c) * __cosf(lat_n) * __cosf(lon_n - lon_c);
        cosd = fminf(fmaxf(cosd, -1.0f + 1e-7f), 1.0f - 1e-7f);
        dS[c * 8 + h] = acosf(cosd);
        mS[c * 8 + h] = (float)nh_mask[n * NHN + h];
        iS[c * 8 + h] = m;
    }

    // ---- stage 0b: normalized Gaussian weights ndT[c][os][h] (K-major pairs) ----
    {
        const int os = lane16;
        const float sg  = sigma[os & 3];
        const float off = dist_offsets[os >> 2];
        const float inv_sg = 1.0f / sg;
        float ndv[NHN];
        float sum = 0.0f;
        #pragma unroll
        for (int h = 0; h < NHN; ++h) {
            const float d = dS[c * 8 + h];
            const float t = (d - off) * inv_sg;
            const float val = __expf(-0.5f * t * t) * mS[c * 8 + h];
            ndv[h] = val;
            sum += val;
        }
        const float inv = 1.0f / (sum + 1e-8f);
        float* row = ndP + c * 160 + os * ND_STRIDE;
        #pragma unroll
        for (int h = 0; h < NHN; ++h) row[h] = ndv[h] * inv;
        row[7] = 0.0f;                               // K padding
    }

    // ---- stage 0c: gather neighbor features; write (h0,h1) pairs as b64 ----
    {
        const int b  = lane >> 4;
        const int v  = (lane >> 2) & 3;
        const int e  = (lane & 3) * 4;
        const int col0 = lane * 4;                   // bve = b*64 + v*16 + e
        const size_t xoff = (size_t)b * NC * 64 + v * 16 + e;
        #pragma unroll
        for (int c2 = 0; c2 < 2; ++c2) {
            #pragma unroll
            for (int hp = 0; hp < 4; ++hp) {         // h pair (2hp, 2hp+1)
                const int m0 = __builtin_amdgcn_readfirstlane(iS[c2 * 8 + 2 * hp]);
                const float4 v0 = *(const float4*)(x + xoff + (size_t)m0 * 64);
                float4 v1 = make_float4(0.f, 0.f, 0.f, 0.f);
                if (hp < 3) {                        // h=7 stays zero pad
                    const int m1 = __builtin_amdgcn_readfirstlane(iS[c2 * 8 + 2 * hp + 1]);
                    v1 = *(const float4*)(x + xoff + (size_t)m1 * 64);
                }
                v2f* bp = (v2f*)(xgP + c2 * 1024 + hp * 256 + col0 * 2);
                bp[0] = (v2f){v0.x, v1.x};           // one ds_store_b64 per column
                bp[1] = (v2f){v0.y, v1.y};
                bp[2] = (v2f){v0.z, v1.z};
                bp[3] = (v2f){v0.w, v1.w};
            }
        }
    }

    __syncthreads();   // covers W-transpose visibility; per-wave data ordered by DScnt

    // ---- stage 1: Y[c] = ND[c](16x8) @ Xg[c](8x128), wmma f32 16x16x4 ----
    #pragma unroll
    for (int c2 = 0; c2 < 2; ++c2) {
        // hoist the 2 A-pairs (K chunks) for this cell into registers
        v2f Ak[2];
        #pragma unroll
        for (int kk = 0; kk < 2; ++kk)
            Ak[kk] = *(const v2f*)(ndP + c2 * 160 + lane16 * ND_STRIDE
                                   + kk * 4 + hi * 2);
        #pragma unroll
        for (int t = 0; t < 8; ++t) {
            v8f acc = {0.f, 0.f, 0.f, 0.f, 0.f, 0.f, 0.f, 0.f};
            #pragma unroll
            for (int kk = 0; kk < 2; ++kk) {
                const v2f Bm = *(const v2f*)(xgP + c2 * 1024 + (kk * 2 + hi) * 256
                                             + (t * 16 + lane16) * 2);
                acc = __builtin_amdgcn_wmma_f32_16x16x4_f32(
                        false, Ak[kk], false, Bm, (short)0, acc, false, false);
            }
            // C vgpr j -> row os = j + hi*8 ; write with skew 2*t
            float* yrow = yP + c2 * Y_CSTRIDE + t * 18 + lane16;
            #pragma unroll
            for (int j = 0; j < 8; ++j)
                yrow[(j + hi * 8) * Y_RSTRIDE] = acc[j];
        }
    }

    // ---- stage 2: out[(c,b,v), f] = Y(16x256) @ W(256x16) + b_out ----
    {
        const float bias = b_out[lane16];
        v8f acc = {bias, bias, bias, bias, bias, bias, bias, bias};

        const int r  = lane16;                        // row = (c,b,v)
        const int cA = r >> 3;
        const int bA = (r >> 2) & 1;
        const int vA = r & 3;
        const int abase = cA * Y_CSTRIDE + bA * 64 + vA * 16 + (bA * 4 + vA) * 2;

        #pragma unroll
        for (int kk = 0; kk < 64; ++kk) {             // K = 256 (os,e)
            const int k0 = kk * 4 + hi * 2;           // even -> contiguous aligned pairs
            const v2f A  = *(const v2f*)(yP + abase + (k0 >> 4) * Y_RSTRIDE + (k0 & 15));
            const v2f Bm = *(const v2f*)(wS + lane16 * W_STRIDE + k0);
            acc = __builtin_amdgcn_wmma_f32_16x16x4_f32(
                    false, A, false, Bm, (short)0, acc, false, false);
        }

        #pragma unroll
        for (int j = 0; j < 8; ++j) {                 // C vgpr j -> row j + hi*8
            const int row = j + hi * 8;
            const int cO  = row >> 3;
            const int bO  = (row >> 2) & 1;
            const int vO  = row & 3;
            const int n   = n0 + cO;
            out[(((size_t)bO * NC + n) * NVV + vO) * EOUT + lane16] = acc[j];
        }
    }
}

extern "C" void kernel_launch(void* const* d_in, const int* in_sizes, int n_in,
                              void* d_out, int out_size, void* d_ws, size_t ws_size,
                              hipStream_t stream) {
    (void)in_sizes; (void)n_in; (void)out_size; (void)d_ws; (void)ws_size;
    const float* x            = (const float*)d_in[0];
    const float* coords       = (const float*)d_in[1];
    const float* sigma        = (const float*)d_in[2];
    const float* dist_offsets = (const float*)d_in[3];
    const float* Wg           = (const float*)d_in[4];
    const float* b_out        = (const float*)d_in[5];
    const int*   adjc         = (const int*)d_in[6];
    const int*   nh_mask      = (const int*)d_in[7];
    float*       out          = (float*)d_out;

    const int blocks = NC / 16;                        // 8 waves * 2 cells per block
    const size_t shmem = (size_t)LDS_FLOATS * sizeof(float);   // ~254 KB (< 320 KB/WG)
    multi_grid_encoder_kernel<<<blocks, 256, shmem, stream>>>(
        x, coords, sigma, dist_offsets, Wg, b_out, adjc, nh_mask, out);
}